// SupervisedContrastiveLoss_4054449127617
// MI455X (gfx1250) — compile-verified
//
#include <hip/hip_runtime.h>

// Supervised contrastive loss, fused, for MI455X (gfx1250, wave32).
// sim = (f fT)/T via V_WMMA_F32_16X16X4_F32, never materialized.
// B tiles are staged global->LDS with CDNA5 async-to-LDS ops (double buffered),
// shared by all 8 waves of the workgroup to cut L2 traffic ~8x.

typedef float v2f __attribute__((ext_vector_type(2)));
typedef float v8f __attribute__((ext_vector_type(8)));

#define N_ROWS 8192
#define DIMS   512
#define TEMP   0.07f
#define INV_T  (1.0f / TEMP)
#define BM     64                      // rows per workgroup (4 strips of 16)
#define THREADS 256                    // 8 waves: 4 strips x 2 column halves
#define LDS_STRIDE 516                 // 516 % 64 = 4 -> conflict-free b64 reads
#define A_FLOATS (BM * LDS_STRIDE)     // 33024 floats
#define BTILE32  (32 * LDS_STRIDE)     // 16512 floats per 32-col staged buffer
#define LDS_FLOATS (A_FLOATS + 2 * BTILE32)   // 66048 floats = 264192 B < 320 KB

// ---- async global->LDS staging of `rows` x 512 fp32 (row-major) into padded LDS ----
__device__ __forceinline__ void async_copy_rows(const float* __restrict__ src,
                                                float* dst, int rows, int tid) {
  const int total = rows * (DIMS / 4);          // float4 granularity
  for (int i = tid; i < total; i += THREADS) {  // uniform trip count across block
    const int r  = i >> 7;                      // / (DIMS/4)
    const int c4 = i & 127;
    const unsigned long long g = (unsigned long long)(size_t)(src + (size_t)r * DIMS + c4 * 4);
    const unsigned l = (unsigned)(size_t)(dst + r * LDS_STRIDE + c4 * 4); // low 32b = LDS offset
    asm volatile("global_load_async_to_lds_b128 %0, %1, off"
                 :: "v"(l), "v"(g) : "memory");
  }
}

__device__ __forceinline__ void wait_async_and_barrier() {
  asm volatile("s_wait_asynccnt 0x0" ::: "memory");
  __syncthreads();
}

// ---------------- kernel 1: L2 row normalization ----------------
__global__ __launch_bounds__(256) void scl_normalize(const float* __restrict__ feat,
                                                     float* __restrict__ fnorm) {
  const int lane = threadIdx.x & 31;
  const int wave = threadIdx.x >> 5;
  const int row  = blockIdx.x * 8 + wave;          // one wave per row
  const float4* src = (const float4*)(feat + (size_t)row * DIMS);
  float4*       dst = (float4*)(fnorm + (size_t)row * DIMS);
  float ss = 0.f;
  float4 v[4];
#pragma unroll
  for (int j = 0; j < 4; ++j) {
    v[j] = src[lane + 32 * j];
    ss += v[j].x * v[j].x + v[j].y * v[j].y + v[j].z * v[j].z + v[j].w * v[j].w;
  }
#pragma unroll
  for (int off = 16; off; off >>= 1) ss += __shfl_xor(ss, off);
  const float sc = 1.0f / fmaxf(sqrtf(ss), 1e-12f);
#pragma unroll
  for (int j = 0; j < 4; ++j) {
    float4 o = { v[j].x * sc, v[j].y * sc, v[j].z * sc, v[j].w * sc };
    dst[lane + 32 * j] = o;
  }
}

// ---------------- kernel 2: fused Gram + masked log-softmax stats ----------------
__global__ __launch_bounds__(THREADS) void scl_main(const float* __restrict__ fnorm,
                                                    const int*   __restrict__ labels,
                                                    float* __restrict__ rowloss,
                                                    float* __restrict__ rowvalid) {
  extern __shared__ float lds[];
  float* As = lds;                                   // BM x LDS_STRIDE
  float* Bbuf0 = lds + A_FLOATS;                     // 32 x LDS_STRIDE
  float* Bbuf1 = Bbuf0 + BTILE32;

  const int tid  = threadIdx.x;
  const int lane = tid & 31;
  const int wave = tid >> 5;                         // 0..7
  const int strip   = wave & 3;                      // row strip 0..3
  const int colhalf = wave >> 2;                     // 0 or 1: which 16 cols of the 32
  const int blockRow0 = blockIdx.x * BM;

  // prologue: async-stage A block and first 32-col B tile, then wait+barrier
  async_copy_rows(fnorm + (size_t)blockRow0 * DIMS, As, BM, tid);
  async_copy_rows(fnorm, Bbuf0, 32, tid);
  wait_async_and_barrier();

  const int lh = lane & 15;                          // N / M position within tile
  const int hi = lane >> 4;                          // lane half
  const int r0 = blockRow0 + strip * 16;             // this wave's 16-row strip

  // row labels for strip: lane lh holds labels[r0+lh]; broadcast per C-VGPR r
  const int rl = labels[r0 + lh];
  int rowlab[8];
#pragma unroll
  for (int r = 0; r < 8; ++r) {
    const int lo = __shfl(rl, r);
    const int hg = __shfl(rl, r + 8);
    rowlab[r] = hi ? hg : lo;
  }

  float cm[8], es[8], ps[8], cs[8];
#pragma unroll
  for (int r = 0; r < 8; ++r) { cm[r] = -1e30f; es[r] = 0.f; ps[r] = 0.f; cs[r] = 0.f; }

  // fp32 WMMA fragments: lane L -> row/col L%16, K-pair offset 2*(L/16); float2 LDS loads
  const float* abase = &As[(strip * 16 + lh) * LDS_STRIDE + (hi << 1)];
  const float  M = INV_T;                            // stable shift ~= true row max

  for (int c0 = 0; c0 < N_ROWS; c0 += 32) {
    const int   cur  = (c0 >> 5) & 1;
    float* Bcur  = cur ? Bbuf1 : Bbuf0;
    float* Bnext = cur ? Bbuf0 : Bbuf1;
    if (c0 + 32 < N_ROWS)                            // uniform branch
      async_copy_rows(fnorm + (size_t)(c0 + 32) * DIMS, Bnext, 32, tid);

    const int col = c0 + colhalf * 16 + lh;
    const float* bbase = &Bcur[(colhalf * 16 + lh) * LDS_STRIDE + (hi << 1)];
    v8f acc0 = {0,0,0,0,0,0,0,0};
    v8f acc1 = {0,0,0,0,0,0,0,0};
#pragma unroll 8
    for (int k = 0; k < DIMS; k += 8) {              // two independent accumulation chains
      v2f a0 = *(const v2f*)(abase + k);
      v2f b0 = *(const v2f*)(bbase + k);
      acc0 = __builtin_amdgcn_wmma_f32_16x16x4_f32(false, a0, false, b0,
                                                   (short)0, acc0, false, false);
      v2f a1 = *(const v2f*)(abase + k + 4);
      v2f b1 = *(const v2f*)(bbase + k + 4);
      acc1 = __builtin_amdgcn_wmma_f32_16x16x4_f32(false, a1, false, b1,
                                                   (short)0, acc1, false, false);
    }
    const int collab = labels[col];
#pragma unroll
    for (int r = 0; r < 8; ++r) {
      const int row  = r0 + r + (hi << 3);
      const float v  = (acc0[r] + acc1[r]) * INV_T;
      const bool self = (col == row);
      cm[r] = fmaxf(cm[r], v);                       // row max includes self
      es[r] += self ? 0.f : __expf(v - M);           // logits_mask excludes self
      const bool match = (collab == rowlab[r]) && !self;
      ps[r] += match ? v : 0.f;
      cs[r] += match ? 1.f : 0.f;
    }
    wait_async_and_barrier();                        // next buffer ready, all readers done
  }

  // butterfly within 16-lane halves -> stats replicated per half
#pragma unroll
  for (int r = 0; r < 8; ++r) {
#pragma unroll
    for (int off = 8; off; off >>= 1) {
      cm[r] = fmaxf(cm[r], __shfl_xor(cm[r], off));
      es[r] += __shfl_xor(es[r], off);
      ps[r] += __shfl_xor(ps[r], off);
      cs[r] += __shfl_xor(cs[r], off);
    }
  }

  // cross-wave merge: waves 4..7 (column half 1) deposit stats, waves 0..3 combine.
  // Valid because es uses the fixed shift M; partial sums add, maxes merge with fmax.
  float* MRG = lds;                                  // reuse A area after final barrier
  if (colhalf == 1 && lh == 0) {
#pragma unroll
    for (int r = 0; r < 8; ++r) {
      const int base = (((strip * 2 + hi) * 8) + r) * 4;
      MRG[base + 0] = cm[r]; MRG[base + 1] = es[r];
      MRG[base + 2] = ps[r]; MRG[base + 3] = cs[r];
    }
  }
  __syncthreads();
  if (colhalf == 0 && lh == 0) {
#pragma unroll
    for (int r = 0; r < 8; ++r) {
      const int base = (((strip * 2 + hi) * 8) + r) * 4;
      const float cmx = fmaxf(cm[r], MRG[base + 0]);
      const float S0  = es[r] + MRG[base + 1];
      const float P   = ps[r] + MRG[base + 2];
      const float C   = cs[r] + MRG[base + 3];
      const int row = r0 + r + (hi << 3);
      const float S    = S0 * __expf(M - cmx);       // = sum exp(sim - rowmax)
      const float logs = __logf(fmaxf(S, 1e-6f));    // clip as reference
      float mlpp = (C > 0.5f) ? (P / C - cmx - logs) : 0.f;
      mlpp = fminf(fmaxf(mlpp, -50.f), 50.f);
      const float valid = (C > 1.5f) ? 1.f : 0.f;    // safe_mask_sum > 1
      rowloss[row]  = -mlpp * valid;                 // (T/baseT)==1
      rowvalid[row] = valid;
    }
  }
}

// ---------------- kernel 3: final scalar reduction ----------------
__global__ __launch_bounds__(256) void scl_reduce(const float* __restrict__ rowloss,
                                                  const float* __restrict__ rowvalid,
                                                  float* __restrict__ out) {
  __shared__ float sl[256], sv[256];
  const int tid = threadIdx.x;
  float a = 0.f, b = 0.f;
  for (int i = tid; i < N_ROWS; i += 256) { a += rowloss[i]; b += rowvalid[i]; }
  sl[tid] = a; sv[tid] = b;
  __syncthreads();
  for (int s = 128; s; s >>= 1) {
    if (tid < s) { sl[tid] += sl[tid + s]; sv[tid] += sv[tid + s]; }
    __syncthreads();
  }
  if (tid == 0) {
    const float nv = sv[0];
    out[0] = (nv > 0.f) ? (sl[0] / fmaxf(nv, 1.f)) : 0.f;
  }
}

extern "C" void kernel_launch(void* const* d_in, const int* in_sizes, int n_in,
                              void* d_out, int out_size, void* d_ws, size_t ws_size,
                              hipStream_t stream) {
  (void)in_sizes; (void)n_in; (void)out_size; (void)ws_size;
  const float* feat   = (const float*)d_in[0];
  const int*   labels = (const int*)d_in[1];     // integer inputs arrive as int32

  float* fnorm    = (float*)d_ws;                       // 8192*512 floats
  float* rowloss  = fnorm + (size_t)N_ROWS * DIMS;      // 8192 floats
  float* rowvalid = rowloss + N_ROWS;                   // 8192 floats

  scl_normalize<<<N_ROWS / 8, 256, 0, stream>>>(feat, fnorm);

  const size_t ldsBytes = (size_t)LDS_FLOATS * sizeof(float);   // 264192 B
  scl_main<<<N_ROWS / BM, THREADS, ldsBytes, stream>>>(fnorm, labels, rowloss, rowvalid);

  scl_reduce<<<1, 256, 0, stream>>>(rowloss, rowvalid, (float*)d_out);
}